// CameraSequencerBase_77421080478503
// MI455X (gfx1250) — compile-verified
//
#include <hip/hip_runtime.h>

typedef __attribute__((ext_vector_type(2))) float v2f;
typedef __attribute__((ext_vector_type(8))) float v8f;

// SE(3)-exponential camera kernel: two 1->512->3 MLPs, Rodrigues, 4x4 matmul.
// Latency-bound micro-kernel: 1 block x 32 threads (one wave32).
// Second layers (3x512 matvecs, both MLPs fused) run on V_WMMA_F32_16X16X4_F32
// accumulating K=512 in 128 steps — full f32 precision on the matrix pipe.
__global__ __launch_bounds__(32) void se3_exp_mlp_kernel(
    const float* __restrict__ x,      // (4,4)
    const float* __restrict__ t,      // (1,)
    const float* __restrict__ W1,     // (512,1)
    const float* __restrict__ Wb1,    // (512,)
    const float* __restrict__ W2,     // (3,512)
    const float* __restrict__ Wb2,    // (3,)
    const float* __restrict__ V1,     // (512,1)
    const float* __restrict__ Vb1,    // (512,)
    const float* __restrict__ V2,     // (3,512)
    const float* __restrict__ Vb2,    // (3,)
    const float* __restrict__ theta,  // ()
    float* __restrict__ out)          // (4,4)
{
    __shared__ float hsh[2][512];   // hidden activations: [0]=w-branch, [1]=v-branch
    __shared__ float wv6[6];        // w[0..2], v[0..2] after bias
    __shared__ float esh[16];       // exp_i (4x4, row-major)

    const int lane = (int)threadIdx.x;   // 0..31
    const float tv = t[0];

    // ---- Layer 1 (1 -> 512) for both MLPs; stage ReLU outputs in LDS ----
    #pragma unroll
    for (int i = 0; i < 16; ++i) {
        const int idx = lane + 32 * i;
        const float hw = fmaf(W1[idx], tv, Wb1[idx]);
        hsh[0][idx] = fmaxf(hw, 0.0f);
        const float hv = fmaf(V1[idx], tv, Vb1[idx]);
        hsh[1][idx] = fmaxf(hv, 0.0f);
    }
    __syncthreads();

    // ---- Layer 2 (512 -> 3), both MLPs, on the f32 WMMA pipe ----
    // D += A(16x4) x B(4x16), 128 chunks of K=4 (K total = 512).
    //   A rows 0..2 = W2, rows 3..5 = V2, rows 6..15 = 0
    //   B col 0 = h_w chunk, col 1 = h_v chunk, cols 2..15 = 0
    // 32-bit A layout (ISA 7.12.2): lane L<16 holds {A[m=L,k=0],A[m=L,k=1]},
    //                               lane L>=16 holds {A[m=L-16,k=2],A[m=L-16,k=3]}.
    // B uses the same half-wave K striping (rows striped across lanes per VGPR).
    const int m  = lane & 15;
    const int kk = (lane >> 4) << 1;          // 0 or 2
    const float* arow = (m < 3) ? (W2 + m * 512)
                                : ((m < 6) ? (V2 + (m - 3) * 512) : W2);
    const float amask = (m < 6) ? 1.0f : 0.0f;     // zero pad rows 6..15
    const float* brow = (m == 1) ? hsh[1] : hsh[0];
    const float bmask = (m < 2) ? 1.0f : 0.0f;     // zero pad cols 2..15

    v8f acc = {};
    #pragma unroll 4
    for (int c = 0; c < 128; ++c) {
        const int kb = 4 * c + kk;
        v2f a, b;
        a.x = arow[kb]     * amask;   // global_load_b64 per lane
        a.y = arow[kb + 1] * amask;
        b.x = brow[kb]     * bmask;   // ds_load_b64 per lane
        b.y = brow[kb + 1] * bmask;
        acc = __builtin_amdgcn_wmma_f32_16x16x4_f32(
            false, a, false, b, (short)0, acc, false, false);
    }

    // D layout: VGPR r -> rows r (lanes 0-15) and r+8 (lanes 16-31), N = lane%16.
    // w[j] = D[j,0]   -> lane 0, acc[j]
    // v[j] = D[3+j,1] -> lane 1, acc[3+j]
    if (lane == 0) {
        wv6[0] = acc[0] + Wb2[0];
        wv6[1] = acc[1] + Wb2[1];
        wv6[2] = acc[2] + Wb2[2];
    } else if (lane == 1) {
        wv6[3] = acc[3] + Vb2[0];
        wv6[4] = acc[4] + Vb2[1];
        wv6[5] = acc[5] + Vb2[2];
    }
    __syncthreads();

    // ---- Rodrigues / SE(3) exponential: scalar tail on lane 0 ----
    if (lane == 0) {
        const float w0 = wv6[0], w1 = wv6[1], w2 = wv6[2];
        const float v0 = wv6[3], v1 = wv6[4], v2 = wv6[5];
        const float th  = theta[0];
        const float s   = sinf(th);
        const float cth = cosf(th);
        const float omc = 1.0f - cth;   // 1 - cos
        const float tms = th - s;       // theta - sin

        float ss[3][3] = {{0.0f, -w2, w1}, {w2, 0.0f, -w0}, {-w1, w0, 0.0f}};
        float ss2[3][3];
        #pragma unroll
        for (int i = 0; i < 3; ++i) {
            #pragma unroll
            for (int j = 0; j < 3; ++j) {
                ss2[i][j] = ss[i][0]*ss[0][j] + ss[i][1]*ss[1][j] + ss[i][2]*ss[2][j];
            }
        }

        const float vv[3] = {v0, v1, v2};
        #pragma unroll
        for (int i = 0; i < 3; ++i) {
            float tr = 0.0f;
            #pragma unroll
            for (int j = 0; j < 3; ++j) {
                const float e   = (i == j) ? 1.0f : 0.0f;
                const float Rij = e + s * ss[i][j] + omc * ss2[i][j];
                const float Vij = e * th + omc * ss[i][j] + tms * ss2[i][j];
                esh[i * 4 + j] = Rij;
                tr += Vij * vv[j];
            }
            esh[i * 4 + 3] = tr;       // trans = Vmat @ v
        }
        esh[12] = 0.0f; esh[13] = 0.0f; esh[14] = 0.0f; esh[15] = 1.0f;
    }
    __syncthreads();

    // ---- out = exp_i @ x : one output element per lane (16 lanes) ----
    if (lane < 16) {
        const int r  = lane >> 2;
        const int cc = lane & 3;
        const float o = esh[r*4 + 0] * x[cc]
                      + esh[r*4 + 1] * x[4  + cc]
                      + esh[r*4 + 2] * x[8  + cc]
                      + esh[r*4 + 3] * x[12 + cc];
        out[lane] = o;
    }
}

extern "C" void kernel_launch(void* const* d_in, const int* in_sizes, int n_in,
                              void* d_out, int out_size, void* d_ws, size_t ws_size,
                              hipStream_t stream) {
    (void)in_sizes; (void)n_in; (void)out_size; (void)d_ws; (void)ws_size;
    se3_exp_mlp_kernel<<<1, 32, 0, stream>>>(
        (const float*)d_in[0],   // x
        (const float*)d_in[1],   // t
        (const float*)d_in[2],   // W1
        (const float*)d_in[3],   // Wb1
        (const float*)d_in[4],   // W2
        (const float*)d_in[5],   // Wb2
        (const float*)d_in[6],   // V1
        (const float*)d_in[7],   // Vb1
        (const float*)d_in[8],   // V2
        (const float*)d_in[9],   // Vb2
        (const float*)d_in[10],  // theta
        (float*)d_out);
}